// BBox_58033598104166
// MI455X (gfx1250) — compile-verified
//
#include <hip/hip_runtime.h>
#include <stdint.h>

// ---------------------------------------------------------------------------
// YOLO decode + soft-NMS for MI455X (gfx1250, wave32).
//
//  Stage 1 (decode_kernel, wide):  per level, stage 128 boxes x 85 floats into
//    LDS with one TDM tensor_load_to_lds descriptor per block, decode
//    sigmoid/exp, compute box corners with V_WMMA_F32_16X16X4_F32
//    (corners = [cx,cy,w,h] x T, a genuine 16x4 * 4x16 matmul), write an AoS
//    float4 box array + {score,class} arrays into d_ws.
//  Stage 2 (nms_kernel, 1 block x 1024 threads = 32 waves): 100 sequential
//    soft-NMS rounds, each = fused (decay + running argmax) pass over the
//    L2-resident arrays (1x b128 + 1x b32 load + 1 store per box), wave32
//    shfl + LDS block reductions, early exit once max score <= threshold.
// ---------------------------------------------------------------------------

#define N_TOTAL   100800
#define MAXB      100
#define IOU_THR   0.5f
#define SCORE_THR 0.1f
#define NEG_INF_F (-1000000000.0f)

typedef float    v2f __attribute__((ext_vector_type(2)));
typedef float    v8f __attribute__((ext_vector_type(8)));
typedef unsigned v4u __attribute__((ext_vector_type(4)));
typedef int      v4i __attribute__((ext_vector_type(4)));
typedef int      v8i __attribute__((ext_vector_type(8)));

#define DEC_THREADS 256
#define BOXES_PER_BLOCK 128           // 8 waves x 16 boxes
#define PRED_FLOATS (BOXES_PER_BLOCK * 85)   // 10880 floats = 43.5 KB LDS

__device__ __forceinline__ float sigmoidf_(float x) {
  return 1.0f / (1.0f + __expf(-x));
}

__global__ __launch_bounds__(DEC_THREADS)
void decode_kernel(const float* __restrict__ pred,
                   const float* __restrict__ anchors,
                   int g, int nboxes, int outBase,
                   float4* __restrict__ bb,
                   float* __restrict__ sc, float* __restrict__ cl)
{
  __shared__ float sPred[PRED_FLOATS];
  __shared__ float sCoord[4][BOXES_PER_BLOCK];

  const int tid  = threadIdx.x;
  const int lane = tid & 31;
  const int wave = tid >> 5;
  const int blockStart = blockIdx.x * BOXES_PER_BLOCK;
  const int remBoxes   = nboxes - blockStart;           // may exceed 128
  const float* predG   = pred + (size_t)blockStart * 85;

#if defined(__HIP_DEVICE_COMPILE__) && defined(__gfx1250__)
  // ---- TDM: one 1-D tensor descriptor stages the whole 128x85 slab ----
  if (__builtin_amdgcn_readfirstlane(wave) == 0) {
    uint64_t gaddr = (uint64_t)(uintptr_t)predG;
    uint32_t laddr = (uint32_t)(uintptr_t)(&sPred[0]);  // LDS offset = addr[31:0]
    uint32_t td0   = (uint32_t)remBoxes * 85u;          // tensor_dim0: OOB -> zero-fill
    uint32_t tile0 = (uint32_t)PRED_FLOATS;             // tile_dim0 (16-bit field)

    v4u g0; v8i g1;
    v4i g2 = {0, 0, 0, 0};
    v4i g3 = {0, 0, 0, 0};
    v8i g4 = {0, 0, 0, 0, 0, 0, 0, 0};                   // unused extra group
    g0[0] = 1u;                                          // count=1, user descriptor
    g0[1] = laddr;                                       // lds_addr
    g0[2] = (uint32_t)(gaddr & 0xffffffffu);             // global_addr[31:0]
    g0[3] = (uint32_t)((gaddr >> 32) & 0x1ffffffu)       // global_addr[56:32]
          | (2u << 30);                                  // type = 2 ("image")
    g1[0] = (int)(2u << 16);                             // data_size = 4 bytes
    g1[1] = (int)((td0 & 0xffffu) << 16);                // tensor_dim0[15:0]
    g1[2] = (int)((td0 >> 16) | (1u << 16));             // tensor_dim0[31:16] | tensor_dim1=1
    g1[3] = (int)((tile0 & 0xffffu) << 16);              // tile_dim0
    g1[4] = 0;                                           // tile_dim1=0 (1-D), tile_dim2=0
    g1[5] = (int)td0;                                    // tensor_dim0_stride[31:0]
    g1[6] = 0;
    g1[7] = 0;
    __builtin_amdgcn_tensor_load_to_lds(g0, g1, g2, g3, g4, 0);
    __builtin_amdgcn_s_wait_tensorcnt(0);
  }
#else
  // host-pass / non-gfx1250 fallback (never executed on device here)
  for (int i = tid; i < PRED_FLOATS; i += DEC_THREADS)
    sPred[i] = (i < remBoxes * 85) ? predG[i] : 0.0f;
#endif
  __syncthreads();

  // ---- per-wave: 16 boxes. lanes 0-15 supply (cx,cy) = A[:,0:2],
  //      lanes 16-31 supply (w,h) = A[:,2:4]  (32-bit 16x4 A layout) ----
  const int mb         = lane & 15;
  const int boxInBlock = wave * 16 + mb;
  const int ib         = blockStart + boxInBlock;        // in-level box id
  const bool inRange   = boxInBlock < remBoxes;
  const float* bp      = &sPred[boxInBlock * 85];

  const int a    = ib % 3;
  const int cell = ib / 3;
  const int gx   = cell % g;
  const int gy   = cell / g;
  const float gi = 1.0f / (float)g;

  v2f A;
  if (lane < 16) {
    A.x = (sigmoidf_(bp[0]) + (float)gx) * gi;           // cx
    A.y = (sigmoidf_(bp[1]) + (float)gy) * gi;           // cy
  } else {
    A.x = __expf(bp[2]) * anchors[a * 2 + 0];            // w
    A.y = __expf(bp[3]) * anchors[a * 2 + 1];            // h
  }

  // B = corner transform T (4x16, only columns 0..3 nonzero):
  //   x1 = cx - .5w ; y1 = cy - .5h ; x2 = cx + .5w ; y2 = cy + .5h
  v2f B; B.x = 0.0f; B.y = 0.0f;
  if (lane == 0 || lane == 2)  B.x = 1.0f;               // T[0][0], T[0][2]
  if (lane == 16)              B.x = -0.5f;              // T[2][0]
  if (lane == 18)              B.x =  0.5f;              // T[2][2]
  if (lane == 1 || lane == 3)  B.y = 1.0f;               // T[1][1], T[1][3]
  if (lane == 17)              B.y = -0.5f;              // T[3][1]
  if (lane == 19)              B.y =  0.5f;               // T[3][3]

  v8f C = {};
  v8f D;
#if defined(__HIP_DEVICE_COMPILE__) && defined(__gfx1250__)
  D = __builtin_amdgcn_wmma_f32_16x16x4_f32(false, A, false, B,
                                            (short)0, C, false, false);
#else
  D = C; (void)A; (void)B;                               // host-pass stub
#endif

  // D layout: VGPR r, lane l -> M = r + 8*(l>=16), N = l&15. Coord n of box m
  // lives at lane (n + 16*(m>=8)), VGPR (m&7). Scatter through LDS -> AoS.
  {
    const int cc   = lane & 15;
    const int half = lane >> 4;
    if (cc < 4) {
#pragma unroll
      for (int r = 0; r < 8; ++r)
        sCoord[cc][wave * 16 + r + 8 * half] = D[r];
    }
  }

  // ---- score / class: obj*max(sigmoid(cls)) = sig(obj)*sig(max logit).
  //      lanes l and l+16 split the 80 class logits of box (wave*16 + l&15).
  {
    float bestv = -3.4e38f; int besti = 0;
    const int cbase = (lane < 16) ? 0 : 40;
#pragma unroll 4
    for (int k = 0; k < 40; ++k) {
      float v = bp[5 + cbase + k];
      if (v > bestv) { bestv = v; besti = cbase + k; }   // strict: first max wins
    }
    float ov = __shfl_xor(bestv, 16);
    int   oi = __shfl_xor(besti, 16);
    if (ov > bestv || (ov == bestv && oi < besti)) { bestv = ov; besti = oi; }
    if (lane < 16 && inRange) {
      const int go = outBase + ib;
      sc[go] = sigmoidf_(bp[4]) * sigmoidf_(bestv);
      cl[go] = (float)besti;
    }
  }

  __syncthreads();
  // coalesced 16-byte AoS box stores (global_store_b128)
  for (int idx = tid; idx < BOXES_PER_BLOCK; idx += DEC_THREADS) {
    if (idx < remBoxes) {
      bb[outBase + blockStart + idx] =
          make_float4(sCoord[0][idx], sCoord[1][idx], sCoord[2][idx], sCoord[3][idx]);
    }
  }
}

// ---------------------------------------------------------------------------

#define NMS_THREADS 1024   // 32 waves (wave32)

__device__ __forceinline__ void amax2(float& bv, int& bi, float v, int i) {
  if (v > bv || (v == bv && i < bi)) { bv = v; bi = i; }
}

__device__ __forceinline__ void blockArgmax(float& v, int& i,
                                            float* sV, int* sI, int tid) {
  const int lane = tid & 31, wave = tid >> 5;
#pragma unroll
  for (int off = 16; off; off >>= 1) {
    float ov = __shfl_down(v, off);
    int   oi = __shfl_down(i, off);
    amax2(v, i, ov, oi);
  }
  if (lane == 0) { sV[wave] = v; sI[wave] = i; }
  __syncthreads();
  if (wave == 0) {
    v = sV[lane]; i = sI[lane];                 // exactly 32 waves
#pragma unroll
    for (int off = 16; off; off >>= 1) {
      float ov = __shfl_down(v, off);
      int   oi = __shfl_down(i, off);
      amax2(v, i, ov, oi);
    }
    if (lane == 0) { sV[0] = v; sI[0] = i; }
  }
  __syncthreads();
  v = sV[0]; i = sI[0];
  __syncthreads();                              // allow sV/sI reuse next call
}

__global__ __launch_bounds__(NMS_THREADS)
void nms_kernel(const float4* __restrict__ bb,
                float* __restrict__ sc, const float* __restrict__ cl,
                float* __restrict__ out, int N)
{
  __shared__ float sRedV[32];
  __shared__ int   sRedI[32];
  __shared__ float sBcast[8];
  __shared__ int   sSelIdx[MAXB];
  __shared__ float sSelSc[MAXB];

  const int tid = threadIdx.x;
  if (tid < MAXB) { sSelIdx[tid] = 0; sSelSc[tid] = 0.0f; }
  __syncthreads();

  // initial argmax over scores
  float curV = -3.4e38f; int curJ = 0;
  for (int b = tid; b < N; b += NMS_THREADS) amax2(curV, curJ, sc[b], b);
  blockArgmax(curV, curJ, sRedV, sRedI, tid);

  int it = 0;
  for (; it < MAXB; ++it) {
    if (!(curV > SCORE_THR)) break;             // uniform: state frozen hereafter
    if (tid == 0) {
      sSelIdx[it] = curJ; sSelSc[it] = curV;
      float4 bj = bb[curJ];
      sBcast[0] = bj.x; sBcast[1] = bj.y; sBcast[2] = bj.z; sBcast[3] = bj.w;
      sBcast[4] = (bj.z - bj.x) * (bj.w - bj.y);
    }
    __syncthreads();
    const float jx1 = sBcast[0], jy1 = sBcast[1];
    const float jx2 = sBcast[2], jy2 = sBcast[3], ja = sBcast[4];

    // fused decay + running argmax for the next round
    float bv = -3.4e38f; int bi = 0;
    for (int b = tid; b < N; b += NMS_THREADS) {
      const float4 bx = bb[b];                  // 1x global_load_b128
      const float s   = sc[b];
      const float iw  = fminf(jx2, bx.z) - fmaxf(jx1, bx.x);
      const float ih  = fminf(jy2, bx.w) - fmaxf(jy1, bx.y);
      const float inter = fmaxf(iw, 0.0f) * fmaxf(ih, 0.0f);
      const float ab  = (bx.z - bx.x) * (bx.w - bx.y);
      const float iou = inter / (ja + ab - inter + 1e-9f);
      const float w   = (iou <= IOU_THR) ? __expf(-iou * iou) : 0.0f; // -0.5/sigma = -1
      const float ns  = (b == curJ) ? NEG_INF_F : s * w;
      sc[b] = ns;
      amax2(bv, bi, ns, b);
    }
    __syncthreads();
    curV = bv; curJ = bi;
    blockArgmax(curV, curJ, sRedV, sRedI, tid);
  }

  __syncthreads();
  // outputs: boxes[100*4] | scores[100] | classes[100] | valid[1]
  if (tid < MAXB) {
    const int j = sSelIdx[tid];
    const float4 bj = bb[j];
    out[tid * 4 + 0] = bj.x;
    out[tid * 4 + 1] = bj.y;
    out[tid * 4 + 2] = bj.z;
    out[tid * 4 + 3] = bj.w;
    out[4 * MAXB + tid] = sSelSc[tid];
    out[5 * MAXB + tid] = cl[j];
  }
  if (tid == 0) out[6 * MAXB] = (float)it;      // num_valid
}

// ---------------------------------------------------------------------------

extern "C" void kernel_launch(void* const* d_in, const int* in_sizes, int n_in,
                              void* d_out, int out_size, void* d_ws, size_t ws_size,
                              hipStream_t stream)
{
  const float* pred0 = (const float*)d_in[0];
  const float* pred1 = (const float*)d_in[1];
  const float* pred2 = (const float*)d_in[2];
  const float* an0   = (const float*)d_in[3];
  const float* an1   = (const float*)d_in[4];
  const float* an2   = (const float*)d_in[5];

  const int N = N_TOTAL;
  float*  ws = (float*)d_ws;          // 6*N floats = 2.42 MB of scratch
  float4* bb = (float4*)d_ws;         // [N] AoS boxes (16B aligned)
  float*  sc = ws + (size_t)4 * N;
  float*  cl = ws + (size_t)5 * N;

  // level 0: g=160 -> 76800 boxes ; level 1: g=80 -> 19200 ; level 2: g=40 -> 4800
  decode_kernel<<<(76800 + 127) / 128, DEC_THREADS, 0, stream>>>(
      pred0, an0, 160, 76800, 0,     bb, sc, cl);
  decode_kernel<<<(19200 + 127) / 128, DEC_THREADS, 0, stream>>>(
      pred1, an1, 80,  19200, 76800, bb, sc, cl);
  decode_kernel<<<(4800 + 127) / 128, DEC_THREADS, 0, stream>>>(
      pred2, an2, 40,  4800,  96000, bb, sc, cl);

  nms_kernel<<<1, NMS_THREADS, 0, stream>>>(bb, sc, cl, (float*)d_out, N);
}